// RNN_torch_75617194214072
// MI455X (gfx1250) — compile-verified
//
#include <hip/hip_runtime.h>
#include <math.h>

// Problem constants (from reference)
constexpr int kN  = 1024;
constexpr int kT  = 512;
constexpr int kB  = 128;
constexpr int kNI = 6;
constexpr int kNO = 2;
constexpr int kTB = kT * kB;            // 65536
constexpr float kAlpha = 0.1f;
constexpr float kGamma = 0.1f;
constexpr float kD     = 0.0f;
constexpr float kSlope = 1.0f;
// sqrt(2/alpha) * sigma = sqrt(20) * 0.03
constexpr float kInpScale = 0.13416407864998739f;
constexpr float kRecScale = 0.13416407864998739f;

constexpr int kBlocks    = 64;          // 8 groups (b-tiles) x 8 blocks
constexpr int kThreads   = 256;         // 8 waves/block; one 16x16 tile per wave
constexpr int kGrpBlocks = 8;           // blocks per b-tile group (barrier members)
constexpr int kBarStride = 32;          // u32 words per group barrier (128B pad)

typedef float v2f __attribute__((ext_vector_type(2)));
typedef float v4f __attribute__((ext_vector_type(4)));
typedef float v8f __attribute__((ext_vector_type(8)));

// ---------------------------------------------------------------------------
// Init: zero outputs + per-group barrier words, broadcast y_init into sbuf[0].
// ---------------------------------------------------------------------------
__global__ void rnn_init_kernel(float* __restrict__ out, float* __restrict__ sbuf,
                                const float* __restrict__ yinit,
                                unsigned* __restrict__ bar) {
  int idx = blockIdx.x * blockDim.x + threadIdx.x;        // 0 .. 131071
  if (idx < kNO * kTB) out[idx] = 0.0f;
  if (idx < kN * kB)   sbuf[idx] = yinit[idx >> 7];       // s0[n,b] = y_init[n]
  if (idx < 8 * kBarStride) bar[idx] = 0u;
}

// outputs[:,0,:] = W_out @ s0  (s0 columns all equal y_init)
__global__ void rnn_out0_kernel(float* __restrict__ out,
                                const float* __restrict__ Wout,
                                const float* __restrict__ yinit) {
  int tid = threadIdx.x;                 // 256 threads: o = tid/128, b = tid%128
  int o = tid >> 7;
  int b = tid & 127;
  float acc = 0.0f;
  for (int n = 0; n < kN; ++n) acc += Wout[o * kN + n] * yinit[n];
  out[o * kTB + b] = acc;                // t = 0 row
}

// ---------------------------------------------------------------------------
// Persistent time-stepping kernel.
//   The recurrence is independent per batch column: group g = b-tile (16 cols),
//   8 blocks/group, 8 waves/block -> 64 waves/group, one 16x16 tile per wave.
//   Per-step sync is a group-local 8-block barrier (not device-wide).
//   fp32 WMMA 16x16x4 for the recurrent GEMM; Euler update + W_out fused.
// ---------------------------------------------------------------------------
__launch_bounds__(kThreads, 1)
__global__ void rnn_persistent_kernel(const float* __restrict__ u,
                                      const float* __restrict__ rnoise,
                                      const float* __restrict__ inoise,
                                      const float* __restrict__ Wrec,
                                      const float* __restrict__ Winp,
                                      const float* __restrict__ Wout,
                                      float* __restrict__ out,
                                      float* __restrict__ sbuf,
                                      unsigned* __restrict__ bar) {
  __shared__ float lds[2][64 * 16];                    // double-buffered s tile

  const int tid  = threadIdx.x;
  const int wid  = tid >> 5;
  const int lane = tid & 31;
  const int hi   = lane >> 4;                          // A/B half: K pair select
  const int lo   = lane & 15;

  const int mt = (blockIdx.x & 7) * 8 + wid;           // 0..63 (m tile)
  const int bt = blockIdx.x >> 3;                      // 0..7  (b tile == group)
  const int m0 = mt * 16;
  const int b0 = bt * 16;
  const int b  = b0 + lo;                              // this lane's batch column

  // A tile source: lane holds W_rec[m0+lo, k + 2*hi .. +1] (contiguous 8B load)
  const float* Arow = Wrec + (size_t)(m0 + lo) * kN + hi * 2;

  // Staging assignment: thread moves one 16B vector of the 64x16 chunk.
  const int srow = tid >> 2;                           // 0..63
  const int scol = (tid & 3) * 4;                      // 0,4,8,12

  unsigned* cnt = bar + bt * kBarStride;               // group-local barrier
  unsigned* gen = cnt + 1;

  #pragma unroll 1
  for (int t = 0; t < kT - 1; ++t) {
    const float* scur = sbuf + (size_t)(t & 1) * (kN * kB);
    float*       snxt = sbuf + (size_t)((t + 1) & 1) * (kN * kB);

    // ---- issue all independent loads early (overlap with the GEMM) ----
    float x[kNI];
    #pragma unroll
    for (int j = 0; j < kNI; ++j) {
      size_t off = (size_t)j * kTB + (size_t)t * kB + b;
      x[j] = u[off] + kInpScale * inoise[off];
    }
    float rt[8], so[8];
    #pragma unroll
    for (int r = 0; r < 8; ++r) {
      const int m = m0 + r + 8 * hi;
      rt[r] = rnoise[(size_t)m * kTB + (size_t)t * kB + b];   // HBM (only HBM read)
      so[r] = scur[(size_t)m * kB + b];
    }

    // ---- recurrent GEMM: 16 chunks of K=64, LDS double-buffered ----
    // NOTE: chunk loop must NOT unroll (it would spill the A tiles to scratch).
    v8f acc = {};
    v4f p0 = *(const v4f*)(scur + (size_t)srow * kB + b0 + scol);            // chunk 0
    v4f p1 = *(const v4f*)(scur + (size_t)(64 + srow) * kB + b0 + scol);     // chunk 1
    #pragma unroll 1
    for (int c = 0; c < 16; ++c) {
      __syncthreads();                                 // buf[c&1] free again
      *(v4f*)&lds[c & 1][srow * 16 + scol] = p0;
      p0 = p1;
      if (c < 14)
        p1 = *(const v4f*)(scur + (size_t)((c + 2) * 64 + srow) * kB + b0 + scol);
      __syncthreads();                                 // staged tile visible
      const float* lb = lds[c & 1];
      const int kb = c * 64;
      #pragma unroll
      for (int kk = 0; kk < 64; kk += 4) {
        v2f a = *(const v2f*)(Arow + kb + kk);         // A: 16x4 of W_rec
        v2f bm;                                        // B: 4x16 of s (LDS)
        bm.x = lb[(kk + hi * 2) * 16 + lo];
        bm.y = lb[(kk + hi * 2 + 1) * 16 + lo];
        acc = __builtin_amdgcn_wmma_f32_16x16x4_f32(
            false, a, false, bm, (short)0, acc, false, false);
      }
    }

    // ---- fused Euler update + output-projection partials ----
    float po0 = 0.0f, po1 = 0.0f;
    #pragma unroll
    for (int r = 0; r < 8; ++r) {
      const int m = m0 + r + 8 * hi;                   // C/D layout: M = r + 8*hi
      float h = acc[r];
      #pragma unroll
      for (int j = 0; j < kNI; ++j) h += Winp[m * kNI + j] * x[j];
      float act = fmaxf(0.0f, kSlope * h);
      float s   = so[r];
      float rhs = -(s - kD) + act + kRecScale * rt[r] - kGamma * s * s * s;
      float sn  = s + kAlpha * rhs;
      snxt[(size_t)m * kB + b] = sn;
      po0 += Wout[m] * sn;                             // W_out[0, m]
      po1 += Wout[kN + m] * sn;                        // W_out[1, m]
    }
    po0 += __shfl_xor(po0, 16);                        // join half-wave row groups
    po1 += __shfl_xor(po1, 16);
    if (hi == 0) {
      atomicAdd(&out[(size_t)0 * kTB + (size_t)(t + 1) * kB + b], po0);
      atomicAdd(&out[(size_t)1 * kTB + (size_t)(t + 1) * kB + b], po1);
    }

    // ---- group-local barrier (8 blocks, generation counter, agent scope) ----
    __syncthreads();
    if (tid == 0) {
      unsigned g = __hip_atomic_load(gen, __ATOMIC_RELAXED, __HIP_MEMORY_SCOPE_AGENT);
      unsigned prev = __hip_atomic_fetch_add(cnt, 1u, __ATOMIC_ACQ_REL,
                                             __HIP_MEMORY_SCOPE_AGENT);
      if (prev == (unsigned)kGrpBlocks - 1u) {
        __hip_atomic_store(cnt, 0u, __ATOMIC_RELAXED, __HIP_MEMORY_SCOPE_AGENT);
        __hip_atomic_fetch_add(gen, 1u, __ATOMIC_RELEASE, __HIP_MEMORY_SCOPE_AGENT);
      } else {
        while (__hip_atomic_load(gen, __ATOMIC_ACQUIRE,
                                 __HIP_MEMORY_SCOPE_AGENT) == g) {
          __builtin_amdgcn_s_sleep(2);
        }
      }
    }
    __syncthreads();
  }
}

// ---------------------------------------------------------------------------
extern "C" void kernel_launch(void* const* d_in, const int* in_sizes, int n_in,
                              void* d_out, int out_size, void* d_ws, size_t ws_size,
                              hipStream_t stream) {
  const float* u      = (const float*)d_in[0];  // (NI,T,B)
  const float* rnois  = (const float*)d_in[1];  // (N,T,B)
  const float* inois  = (const float*)d_in[2];  // (NI,T,B)
  const float* Wrec   = (const float*)d_in[3];  // (N,N)
  const float* Winp   = (const float*)d_in[4];  // (N,NI)
  const float* Wout   = (const float*)d_in[5];  // (NO,N)
  const float* yinit  = (const float*)d_in[6];  // (N,)
  float* out = (float*)d_out;                   // (NO,T,B)

  // Workspace: ping-pong state buffers (2 * N * B floats = 1 MB) + barriers.
  float* sbuf   = (float*)d_ws;
  unsigned* bar = (unsigned*)((char*)d_ws + (size_t)2 * kN * kB * sizeof(float));

  rnn_init_kernel<<<(kNO * kTB + 255) / 256, 256, 0, stream>>>(out, sbuf, yinit, bar);
  rnn_out0_kernel<<<1, 256, 0, stream>>>(out, Wout, yinit);
  rnn_persistent_kernel<<<kBlocks, kThreads, 0, stream>>>(
      u, rnois, inois, Wrec, Winp, Wout, out, sbuf, bar);
}